// PropeDotProductAttention_46437186404974
// MI455X (gfx1250) — compile-verified
//
#include <hip/hip_runtime.h>
#include <hip/hip_bf16.h>

// Problem constants (from reference): B=2, H=16, C=4, PX=PY=16, EXTRA=64, HD=64
#define B_      2
#define H_      16
#define C_      4
#define N_      1088            // C*PX*PY + EXTRA
#define D_      64
#define EXTRA_  64
#define TOK_    (B_*H_*N_)      // 34816 tokens per tensor
#define LDK     68              // LDS row pitch (64 + 4 pad) to avoid bank conflicts

typedef float v2f __attribute__((ext_vector_type(2)));
typedef float v8f __attribute__((ext_vector_type(8)));

// Cross-lane xor within 16-lane rows via DPP16 row_xmask (pure VALU, no LDS).
template<int MASK>
static __device__ __forceinline__ float rowxor_f(float x) {
  int xi = __float_as_int(x);
  int r  = __builtin_amdgcn_update_dpp(xi, xi, 0x160 | MASK, 0xf, 0xf, true);
  return __int_as_float(r);
}

// ---------------------------------------------------------------------------
// Kernel 0: build per-camera matrices P = lift(Kn) @ viewmat and
//           P_inv = invert_SE3(viewmat) @ lift(invert_K(Kn)); 8 cams total.
// ---------------------------------------------------------------------------
__global__ void prope_setup_kernel(const float* __restrict__ viewmats,
                                   const float* __restrict__ Ks,
                                   float* __restrict__ wsP,
                                   float* __restrict__ wsPinv) {
  int i = threadIdx.x;
  if (i >= B_ * C_) return;
  const float* VM = viewmats + i * 16;   // [4][4]
  const float* K3 = Ks + i * 9;          // [3][3]

  float fx = K3[0] * (1.0f / 256.0f);
  float fy = K3[4] * (1.0f / 256.0f);
  float cx = K3[2] * (1.0f / 256.0f) - 0.5f;
  float cy = K3[5] * (1.0f / 256.0f) - 0.5f;

  // lift(Kn)
  float L[16] = {fx, 0.f, cx, 0.f,
                 0.f, fy, cy, 0.f,
                 0.f, 0.f, 1.f, 0.f,
                 0.f, 0.f, 0.f, 1.f};
  // P = L @ VM
  float P[16];
  #pragma unroll
  for (int r = 0; r < 4; ++r)
    #pragma unroll
    for (int c = 0; c < 4; ++c) {
      float s = 0.f;
      #pragma unroll
      for (int k = 0; k < 4; ++k) s += L[r*4+k] * VM[k*4+c];
      P[r*4+c] = s;
    }

  // invert_SE3(VM)
  float IV[16];
  IV[0]=VM[0]; IV[1]=VM[4]; IV[2]=VM[8];
  IV[4]=VM[1]; IV[5]=VM[5]; IV[6]=VM[9];
  IV[8]=VM[2]; IV[9]=VM[6]; IV[10]=VM[10];
  IV[3]  = -(IV[0]*VM[3] + IV[1]*VM[7] + IV[2]*VM[11]);
  IV[7]  = -(IV[4]*VM[3] + IV[5]*VM[7] + IV[6]*VM[11]);
  IV[11] = -(IV[8]*VM[3] + IV[9]*VM[7] + IV[10]*VM[11]);
  IV[12]=0.f; IV[13]=0.f; IV[14]=0.f; IV[15]=1.f;

  // lift(invert_K(Kn))
  float IK[16] = {1.f/fx, 0.f, -cx/fx, 0.f,
                  0.f, 1.f/fy, -cy/fy, 0.f,
                  0.f, 0.f, 1.f, 0.f,
                  0.f, 0.f, 0.f, 1.f};
  float PI[16];
  #pragma unroll
  for (int r = 0; r < 4; ++r)
    #pragma unroll
    for (int c = 0; c < 4; ++c) {
      float s = 0.f;
      #pragma unroll
      for (int k = 0; k < 4; ++k) s += IV[r*4+k] * IK[k*4+c];
      PI[r*4+c] = s;
    }

  #pragma unroll
  for (int j = 0; j < 16; ++j) {
    wsP[i*16 + j]    = P[j];
    wsPinv[i*16 + j] = PI[j];
  }
}

// ---------------------------------------------------------------------------
// Kernel 1: forward transform q,k,v -> qt,kt,vt.
// Two threads per token: role 0 = matrix part (dims 0..31),
//                        role 1 = RoPE part  (dims 32..63).
// ---------------------------------------------------------------------------
__global__ __launch_bounds__(256)
void prope_transform_kernel(const float* __restrict__ q,
                            const float* __restrict__ k,
                            const float* __restrict__ v,
                            const float* __restrict__ wsP,
                            const float* __restrict__ wsPinv,
                            float* __restrict__ qt,
                            float* __restrict__ kt,
                            float* __restrict__ vt) {
  int idx = blockIdx.x * blockDim.x + threadIdx.x;
  if (idx >= 3 * TOK_ * 2) return;
  int role   = idx & 1;
  int g      = idx >> 1;
  int tensor = g / TOK_;          // 0=q, 1=k, 2=v
  int t      = g % TOK_;
  int n      = t % N_;
  int b      = t / (H_ * N_);

  const float* src = (tensor == 0 ? q : tensor == 1 ? k : v) + (size_t)t * D_;
  float*       dst = (tensor == 0 ? qt : tensor == 1 ? kt : vt) + (size_t)t * D_;

  bool spatial = (n >= EXTRA_);

  if (role == 0) {
    if (!spatial) {
      #pragma unroll
      for (int j = 0; j < 32; j += 4)
        *(float4*)(dst + j) = *(const float4*)(src + j);
      return;
    }
    int cam = (n - EXTRA_) >> 8;
    const float* Mb = (tensor == 0 ? wsP : wsPinv) + (b * C_ + cam) * 16;
    float M[16];
    if (tensor == 0) {      // Mq = P^T
      #pragma unroll
      for (int r = 0; r < 4; ++r)
        #pragma unroll
        for (int c = 0; c < 4; ++c) M[r*4+c] = Mb[c*4+r];
    } else {
      #pragma unroll
      for (int j = 0; j < 16; ++j) M[j] = Mb[j];
    }
    #pragma unroll
    for (int g4 = 0; g4 < 8; ++g4) {
      float4 p = *(const float4*)(src + g4*4);
      float o0 = M[0]*p.x  + M[1]*p.y  + M[2]*p.z  + M[3]*p.w;
      float o1 = M[4]*p.x  + M[5]*p.y  + M[6]*p.z  + M[7]*p.w;
      float o2 = M[8]*p.x  + M[9]*p.y  + M[10]*p.z + M[11]*p.w;
      float o3 = M[12]*p.x + M[13]*p.y + M[14]*p.z + M[15]*p.w;
      *(float4*)(dst + g4*4) = make_float4(o0, o1, o2, o3);
    }
  } else {
    if (!spatial) {
      #pragma unroll
      for (int j = 32; j < 64; j += 4)
        *(float4*)(dst + j) = *(const float4*)(src + j);
      return;
    }
    int s = n - EXTRA_;
    float px = (float)(s & 15);
    float py = (float)((s >> 4) & 15);
    float ox[32];
    #pragma unroll
    for (int f = 0; f < 8; ++f) {
      float freq = powf(100.0f, -(float)f * 0.125f);
      float ax = px * freq, ay = py * freq;
      float cxv = cosf(ax), sxv = sinf(ax);
      float xa = src[32+f], xb = src[40+f];
      ox[f]      =  cxv*xa + sxv*xb;
      ox[8+f]    = -sxv*xa + cxv*xb;
      float cyv = cosf(ay), syv = sinf(ay);
      float ya = src[48+f], yb = src[56+f];
      ox[16+f]   =  cyv*ya + syv*yb;
      ox[24+f]   = -syv*ya + cyv*yb;
    }
    #pragma unroll
    for (int j = 0; j < 32; j += 4)
      *(float4*)(dst + 32 + j) = make_float4(ox[j], ox[j+1], ox[j+2], ox[j+3]);
  }
}

// ---------------------------------------------------------------------------
// Kernel 2: fused flash attention (f32 WMMA 16x16x4) + output transform.
// Block = 128 threads (4 waves). Each wave owns a 16-row query tile; the
// block cooperatively stages 32-key K/V tiles in LDS. Row sums are computed
// on the matrix pipe (P @ ones -> broadcast row-sum tile); row max uses
// DPP16 row_xmask butterflies (pure VALU).
// ---------------------------------------------------------------------------
__global__ __launch_bounds__(128)
void prope_attn_kernel(const float* __restrict__ qt,
                       const float* __restrict__ kt,
                       const float* __restrict__ vt,
                       const float* __restrict__ wsP,
                       float* __restrict__ out) {
  __shared__ float smem[2 * 32 * LDK + 4 * 16 * 17];   // K | V | P-scratch
  float* smemK = smem;
  float* smemV = smem + 32 * LDK;
  float* smemP = smem + 2 * 32 * LDK;

  int bh   = blockIdx.x / 17;
  int tile = blockIdx.x % 17;
  int b    = bh / H_;

  const float* qbh = qt + (size_t)bh * N_ * D_;
  const float* kbh = kt + (size_t)bh * N_ * D_;
  const float* vbh = vt + (size_t)bh * N_ * D_;

  int tid  = threadIdx.x;
  int w    = tid >> 5;
  int lane = tid & 31;
  int r    = lane & 15;       // row / col within 16-group
  int h    = lane >> 4;       // lane half
  int qbase = tile * 64 + w * 16;

  // --- load Q A-fragments (scaled by 1/sqrt(64)) ---
  v2f aq[16];
  const float* qrow = qbh + (size_t)(qbase + r) * D_ + 2 * h;
  #pragma unroll
  for (int c = 0; c < 16; ++c) {
    float2 qv = *(const float2*)(qrow + 4 * c);
    aq[c].x = qv.x * 0.125f;
    aq[c].y = qv.y * 0.125f;
  }

  v8f o[4], osum;
  #pragma unroll
  for (int vv = 0; vv < 8; ++vv) {
    osum[vv] = 0.0f;
    #pragma unroll
    for (int t = 0; t < 4; ++t) o[t][vv] = 0.0f;
  }

  float m[8];
  #pragma unroll
  for (int vv = 0; vv < 8; ++vv) m[vv] = -INFINITY;

  v2f bone; bone.x = 1.0f; bone.y = 1.0f;   // ones B-fragment for row sums

  int lrow = tid >> 2;            // 0..31 : K/V row to stage
  int lseg = (tid & 3) * 16;      // 16-col segment
  float* pW = smemP + w * 272;    // per-wave 16x17 P scratch

  for (int kb = 0; kb < N_; kb += 32) {
    // --- cooperative K/V tile load into LDS ---
    const float* ksrc = kbh + (size_t)(kb + lrow) * D_ + lseg;
    const float* vsrc = vbh + (size_t)(kb + lrow) * D_ + lseg;
    float* kdst = smemK + lrow * LDK + lseg;
    float* vdst = smemV + lrow * LDK + lseg;
    #pragma unroll
    for (int j = 0; j < 16; j += 4) {
      *(float4*)(kdst + j) = *(const float4*)(ksrc + j);
      *(float4*)(vdst + j) = *(const float4*)(vsrc + j);
    }
    __syncthreads();

    #pragma unroll
    for (int sub = 0; sub < 2; ++sub) {
      int kb2 = sub * 16;

      // --- S = Q . K^T  (16 x wmma f32 16x16x4) ---
      v8f acc;
      #pragma unroll
      for (int vv = 0; vv < 8; ++vv) acc[vv] = 0.0f;
      #pragma unroll
      for (int c = 0; c < 16; ++c) {
        float2 kv = *(const float2*)(smemK + (kb2 + r) * LDK + 4 * c + 2 * h);
        v2f bk; bk.x = kv.x; bk.y = kv.y;
        acc = __builtin_amdgcn_wmma_f32_16x16x4_f32(
            false, aq[c], false, bk, (short)0, acc, false, false);
      }

      // --- online softmax: row max via DPP butterflies, rescale state ---
      float pv[8];
      #pragma unroll
      for (int vv = 0; vv < 8; ++vv) {
        float x = acc[vv];
        float rm = fmaxf(x,  rowxor_f<1>(x));
        rm = fmaxf(rm, rowxor_f<2>(rm));
        rm = fmaxf(rm, rowxor_f<4>(rm));
        rm = fmaxf(rm, rowxor_f<8>(rm));
        float nm = fmaxf(m[vv], rm);
        float sc = __expf(m[vv] - nm);
        m[vv]  = nm;
        pv[vv] = __expf(x - nm);
        osum[vv] *= sc;
        #pragma unroll
        for (int t = 0; t < 4; ++t) o[t][vv] *= sc;
      }

      // --- P through LDS (C-layout -> A-layout transpose) ---
      #pragma unroll
      for (int vv = 0; vv < 8; ++vv)
        pW[(vv + 8 * h) * 17 + r] = pv[vv];

      // --- O += P . V ; row-sum tile osum += P . ones (20 x wmma) ---
      #pragma unroll
      for (int c = 0; c < 4; ++c) {
        v2f ap;
        ap.x = pW[r * 17 + 4 * c + 2 * h];
        ap.y = pW[r * 17 + 4 * c + 2 * h + 1];
        int key = kb2 + 4 * c + 2 * h;
        osum = __builtin_amdgcn_wmma_f32_16x16x4_f32(
            false, ap, false, bone, (short)0, osum, false, false);
        #pragma unroll
        for (int t = 0; t < 4; ++t) {
          v2f bv;
          bv.x = smemV[key * LDK + t * 16 + r];
          bv.y = smemV[(key + 1) * LDK + t * 16 + r];
          o[t] = __builtin_amdgcn_wmma_f32_16x16x4_f32(
              false, ap, false, bv, (short)0, o[t], false, false);
        }
      }
    }
    __syncthreads();
  }

  // --- normalize (osum column is the row sum, already lane-replicated) ---
  #pragma unroll
  for (int vv = 0; vv < 8; ++vv) {
    float inv = 1.0f / osum[vv];
    #pragma unroll
    for (int t = 0; t < 4; ++t) o[t][vv] *= inv;
  }

  // --- spill O tile to LDS (reuse K/V region: 4 waves x 16 x LDK floats) ---
  float* oW = smem + w * (16 * LDK);
  #pragma unroll
  for (int t = 0; t < 4; ++t)
    #pragma unroll
    for (int vv = 0; vv < 8; ++vv)
      oW[(vv + 8 * h) * LDK + t * 16 + r] = o[t][vv];
  __syncthreads();

  // --- fused output transform (Mo matmul + inverse RoPE) ---
  int n = qbase + r;
  const float* orow = oW + r * LDK;
  float* drow = out + ((size_t)bh * N_ + n) * D_;
  bool spatial = (n >= EXTRA_);

  if (h == 0) {
    if (!spatial) {
      #pragma unroll
      for (int j = 0; j < 32; j += 4) {
        float4 p = make_float4(orow[j], orow[j+1], orow[j+2], orow[j+3]);
        *(float4*)(drow + j) = p;
      }
    } else {
      int cam = (n - EXTRA_) >> 8;
      const float* M = wsP + (b * C_ + cam) * 16;
      #pragma unroll
      for (int g4 = 0; g4 < 8; ++g4) {
        float p0 = orow[g4*4+0], p1 = orow[g4*4+1];
        float p2 = orow[g4*4+2], p3 = orow[g4*4+3];
        float o0 = M[0]*p0  + M[1]*p1  + M[2]*p2  + M[3]*p3;
        float o1 = M[4]*p0  + M[5]*p1  + M[6]*p2  + M[7]*p3;
        float o2 = M[8]*p0  + M[9]*p1  + M[10]*p2 + M[11]*p3;
        float o3 = M[12]*p0 + M[13]*p1 + M[14]*p2 + M[15]*p3;
        *(float4*)(drow + g4*4) = make_float4(o0, o1, o2, o3);
      }
    }
  } else {
    if (!spatial) {
      #pragma unroll
      for (int j = 32; j < 64; j += 4) {
        float4 p = make_float4(orow[j], orow[j+1], orow[j+2], orow[j+3]);
        *(float4*)(drow + j) = p;
      }
    } else {
      int s = n - EXTRA_;
      float px = (float)(s & 15);
      float py = (float)((s >> 4) & 15);
      float ox[32];
      #pragma unroll
      for (int f = 0; f < 8; ++f) {
        float freq = powf(100.0f, -(float)f * 0.125f);
        float cxv = cosf(px * freq), sxv = sinf(px * freq);
        float xa = orow[32+f], xb = orow[40+f];
        ox[f]    = cxv*xa - sxv*xb;        // inverse rope
        ox[8+f]  = sxv*xa + cxv*xb;
        float cyv = cosf(py * freq), syv = sinf(py * freq);
        float ya = orow[48+f], yb = orow[56+f];
        ox[16+f] = cyv*ya - syv*yb;
        ox[24+f] = syv*ya + cyv*yb;
      }
      #pragma unroll
      for (int j = 0; j < 32; j += 4)
        *(float4*)(drow + 32 + j) = make_float4(ox[j], ox[j+1], ox[j+2], ox[j+3]);
    }
  }
}

// ---------------------------------------------------------------------------
// Host launcher
// ---------------------------------------------------------------------------
extern "C" void kernel_launch(void* const* d_in, const int* in_sizes, int n_in,
                              void* d_out, int out_size, void* d_ws, size_t ws_size,
                              hipStream_t stream) {
  const float* q        = (const float*)d_in[0];
  const float* k        = (const float*)d_in[1];
  const float* v        = (const float*)d_in[2];
  const float* viewmats = (const float*)d_in[3];
  const float* Ks       = (const float*)d_in[4];

  float* ws     = (float*)d_ws;
  float* wsP    = ws;                 // 8 * 16 floats
  float* wsPinv = ws + 128;           // 8 * 16 floats
  const size_t TEN = (size_t)TOK_ * D_;   // 2,228,224 floats per tensor
  float* qt = ws + 256;
  float* kt = qt + TEN;
  float* vt = kt + TEN;

  prope_setup_kernel<<<1, 32, 0, stream>>>(viewmats, Ks, wsP, wsPinv);

  int tthreads = 3 * TOK_ * 2;                       // 208,896
  int tblocks  = (tthreads + 255) / 256;             // 816
  prope_transform_kernel<<<tblocks, 256, 0, stream>>>(q, k, v, wsP, wsPinv, qt, kt, vt);

  int ablocks = B_ * H_ * (N_ / 64);                 // 544
  prope_attn_kernel<<<ablocks, 128, 0, stream>>>(qt, kt, vt, wsP, (float*)d_out);
}